// SingleChannelInterpolation_9620726743118
// MI455X (gfx1250) — compile-verified
//
#include <hip/hip_runtime.h>
#include <math.h>

typedef __attribute__((ext_vector_type(2))) float v2f;
typedef __attribute__((ext_vector_type(8))) float v8f;

#define T_PTS 128
#define R_PTS 192
#define CDIM  64
#define KAPPA_F 10.0f
#define REF_STEP (192.0f / 191.0f)

__global__ __launch_bounds__(128)
void sci_fused_kernel(const float* __restrict__ x,
                      const float* __restrict__ kern,
                      float* __restrict__ out) {
    __shared__ float s_xt[T_PTS];
    __shared__ float s_d[T_PTS];
    __shared__ float s_lm[T_PTS];   // log(m)
    __shared__ float s_al[T_PTS];   // softplus(kernel)
    __shared__ float s_mx[R_PTS];   // max_t s(t,r)
    __shared__ float s_mx2[R_PTS];  // max_t s2(t,r)

    const int bc  = blockIdx.x;         // 0..2047
    const int b   = bc >> 6;            // / 64
    const int c   = bc & 63;
    const int tid = threadIdx.x;        // 0..127

    // x layout: (B, 3, C, T)
    const size_t baseX = ((size_t)(b * 3 + 0) * CDIM + c) * T_PTS;
    const size_t baseD = ((size_t)(b * 3 + 1) * CDIM + c) * T_PTS;
    const size_t baseM = ((size_t)(b * 3 + 2) * CDIM + c) * T_PTS;

    // ---- Stage per-(b,c) vectors into LDS (blockDim == T_PTS) ----
    {
        const float xt = x[baseX + tid];
        const float dv = x[baseD + tid];
        const float mv = x[baseM + tid];
        const float kv = kern[tid];
        s_xt[tid] = xt;
        s_d[tid]  = dv;
        s_lm[tid] = __logf(mv);
        // softplus, overflow-safe
        s_al[tid] = fmaxf(kv, 0.0f) + log1pf(__expf(-fabsf(kv)));
    }
    __syncthreads();

    // ---- Pass 1: per-r max of both score streams (exact logsumexp) ----
    for (int r = tid; r < R_PTS; r += 128) {
        const float rt = (float)r * REF_STEP;
        float m1 = -3.4e38f, m2 = -3.4e38f;
        for (int t = 0; t < T_PTS; ++t) {
            const float diff = s_d[t] - rt;
            const float an   = s_al[t] * diff * diff;
            const float lm   = s_lm[t];
            m1 = fmaxf(m1, lm - an);
            m2 = fmaxf(m2, lm - KAPPA_F * an);
        }
        s_mx[r]  = m1;
        s_mx2[r] = m2;
    }
    __syncthreads();

    // ---- Pass 2: fused exp -> WMMA f32 contraction over t ----
    const int wave = tid >> 5;          // 0..3
    const int lane = tid & 31;
    const int nsub = lane & 15;         // column within 16-wide r tile
    const int koff = (lane >> 4) << 1;  // this lane covers t-slots {koff, koff+1} of each 4-chunk

    v2f ones; ones.x = 1.0f; ones.y = 1.0f;   // all-ones A: D rows = column sums of B

    for (int tile = wave; tile < (R_PTS / 16); tile += 4) {
        const int   r    = tile * 16 + nsub;
        const float rt   = (float)r * REF_STEP;
        const float mxr  = s_mx[r];
        const float mx2r = s_mx2[r];

        v8f accN  = {0.f,0.f,0.f,0.f,0.f,0.f,0.f,0.f};  // sum e * x_t      (stream 1)
        v8f accD  = {0.f,0.f,0.f,0.f,0.f,0.f,0.f,0.f};  // sum e            (stream 1)
        v8f accN2 = {0.f,0.f,0.f,0.f,0.f,0.f,0.f,0.f};  // sum e2 * x_t     (stream 2)
        v8f accD2 = {0.f,0.f,0.f,0.f,0.f,0.f,0.f,0.f};  // sum e2           (stream 2)

        #pragma unroll 4
        for (int t0 = 0; t0 < T_PTS; t0 += 4) {
            const int ta = t0 + koff;
            const int tb = ta + 1;

            const float da  = s_d[ta] - rt;
            const float db  = s_d[tb] - rt;
            const float ana = s_al[ta] * da * da;
            const float anb = s_al[tb] * db * db;
            const float lma = s_lm[ta];
            const float lmb = s_lm[tb];

            const float ea  = __expf(lma - ana - mxr);
            const float eb  = __expf(lmb - anb - mxr);
            const float e2a = __expf(lma - KAPPA_F * ana - mx2r);
            const float e2b = __expf(lmb - KAPPA_F * anb - mx2r);

            const float xa = s_xt[ta];
            const float xb = s_xt[tb];

            v2f fE;  fE.x  = ea;       fE.y  = eb;
            v2f fP;  fP.x  = ea * xa;  fP.y  = eb * xb;
            v2f fE2; fE2.x = e2a;      fE2.y = e2b;
            v2f fP2; fP2.x = e2a * xa; fP2.y = e2b * xb;

            accN  = __builtin_amdgcn_wmma_f32_16x16x4_f32(false, ones, false, fP,  (short)0, accN,  false, false);
            accD  = __builtin_amdgcn_wmma_f32_16x16x4_f32(false, ones, false, fE,  (short)0, accD,  false, false);
            accN2 = __builtin_amdgcn_wmma_f32_16x16x4_f32(false, ones, false, fP2, (short)0, accN2, false, false);
            accD2 = __builtin_amdgcn_wmma_f32_16x16x4_f32(false, ones, false, fE2, (short)0, accD2, false, false);
        }

        // D layout: VGPR0 lanes 0-15 hold (M=0, N=lane); with all-ones A every
        // row equals the column sum, so lanes 0-15 read element 0 and store r.
        if (lane < 16) {
            const float num  = accN[0];
            const float den  = accD[0];   // >= 1 (max term contributes exp(0))
            const float num2 = accN2[0];
            const float den2 = accD2[0];

            const size_t oy = ((size_t)(b * 3 + 0) * CDIM + c) * R_PTS + r;
            const size_t ow = ((size_t)(b * 3 + 1) * CDIM + c) * R_PTS + r;
            const size_t ot = ((size_t)(b * 3 + 2) * CDIM + c) * R_PTS + r;

            out[oy] = num / den;
            out[ow] = mxr + __logf(den);
            out[ot] = num2 / den2;
        }
    }
}

extern "C" void kernel_launch(void* const* d_in, const int* in_sizes, int n_in,
                              void* d_out, int out_size, void* d_ws, size_t ws_size,
                              hipStream_t stream) {
    (void)in_sizes; (void)n_in; (void)out_size; (void)d_ws; (void)ws_size;
    const float* x    = (const float*)d_in[0];   // (32, 3, 64, 128) f32
    const float* kern = (const float*)d_in[1];   // (128,) f32
    float* out        = (float*)d_out;           // (32, 3, 64, 192) f32

    sci_fused_kernel<<<dim3(32 * CDIM), dim3(128), 0, stream>>>(x, kern, out);
}